// KalmanNetLayer_6330781794388
// MI455X (gfx1250) — compile-verified
//
#include <hip/hip_runtime.h>
#include <math.h>

// ---------------------------------------------------------------------------
// KalmanNet / gated DeltaNet layer for MI455X (gfx1250, wave32).
//
//   B=4, S=2048, F=512, H=8, D=32, HIDDEN=256, NA=16
//
//  * fp32 WMMA (v_wmma_f32_16x16x4_f32) for the two large GEMMs.
//  * Serial scan in identity+rank-1 form (matvec + 3 rank-1 updates/step).
//  * Scan inputs are prefetched with CDNA5 async global->LDS copies
//    (global_load_async_to_lds_b128, ASYNCcnt, double-buffered 16-step
//    chunks) so the 2048-step dependent chain only ever touches LDS.
//    Uniform-address ds_load_b128 broadcasts replace all cross-lane shuffles.
// ---------------------------------------------------------------------------

typedef __attribute__((ext_vector_type(2))) float v2f;
typedef __attribute__((ext_vector_type(8))) float v8f;

#define B_    4
#define S_    2048
#define F_    512
#define H_    8
#define D_    32
#define HID_  256
#define NA_   16
#define EPS_  1e-6f
#define NTOK  (B_ * S_)          // 8192

// ---- scan chunking ----
#define T_        16             // steps per prefetch chunk
#define NCHUNK    (S_ / T_)      // 128
// per-buffer float offsets inside LDS
#define OFF_K     0
#define OFF_KU    (1 * T_ * D_)  //  512
#define OFF_Q     (2 * T_ * D_)  // 1024
#define OFF_V     (3 * T_ * D_)  // 1536
#define OFF_VU    (4 * T_ * D_)  // 2048
#define OFF_SC    (5 * T_ * D_)  // 2560 : [beta|alpha|gamma|mask] x T_
#define BUF_F     (OFF_SC + 4 * T_)  // 2624 floats = 10496 B per buffer
#define OPS_PER_CHUNK 22         // 20 x b128 + 2 x b32

__device__ __forceinline__ float sigmoidf_(float x) {
    return 1.0f / (1.0f + __expf(-x));
}

__device__ __forceinline__ float waveAllSum(float v) {
    #pragma unroll
    for (int off = 16; off > 0; off >>= 1)
        v += __shfl_xor(v, off, 32);
    return v;
}

// CDNA5 async global->LDS (ASYNCcnt-tracked); inline asm is portable across
// the ROCm-7.2 / amdgpu-toolchain builtin-arity split.
#define ASYNC_B128(ldsoff, gaddr)                                             \
    asm volatile("global_load_async_to_lds_b128 %0, %1, off"                  \
                 :: "v"(ldsoff), "v"(gaddr) : "memory")
#define ASYNC_B32(ldsoff, gaddr)                                              \
    asm volatile("global_load_async_to_lds_b32 %0, %1, off"                   \
                 :: "v"(ldsoff), "v"(gaddr) : "memory")
#define WAIT_ASYNC(n)                                                         \
    asm volatile("s_wait_asynccnt %0" :: "i"(n) : "memory")
#define WAIT_DS0()                                                            \
    asm volatile("s_wait_dscnt 0x0" ::: "memory")

// ---------------------------------------------------------------------------
// Kernel 1: projection GEMM  {qs|ks|vs}_raw = X (8192x512) @ W (512x256)
// One wave per 16x16 tile, fp32 WMMA, K step 4.
// grid = (512, 6), block = 256.
// ---------------------------------------------------------------------------
__global__ void proj_gemm(const float* __restrict__ X,
                          const float* __restrict__ Wq,
                          const float* __restrict__ Wk,
                          const float* __restrict__ Wv,
                          float* __restrict__ qs,
                          float* __restrict__ ks,
                          float* __restrict__ vs)
{
    const int wave  = threadIdx.x >> 5;
    const int lane  = threadIdx.x & 31;
    const int m0    = blockIdx.x * 16;
    const int ntile = blockIdx.y * 8 + wave;         // 0..47
    const int mat   = ntile >> 4;                    // 0:q 1:k 2:v
    const int n0    = (ntile & 15) * 16;

    const float* W = (mat == 0) ? Wq : ((mat == 1) ? Wk : Wv);
    float*       C = (mat == 0) ? qs : ((mat == 1) ? ks : vs);

    const int r     = lane & 15;
    const int khalf = (lane >> 4) * 2;               // lanes 16-31: K+2,K+3

    v8f acc = {};
    const float* Xrow = X + (size_t)(m0 + r) * F_;
    for (int k = 0; k < F_; k += 4) {
        v2f a, b;
        a.x = Xrow[k + khalf];
        a.y = Xrow[k + khalf + 1];
        b.x = W[(size_t)(k + khalf)     * HID_ + n0 + r];
        b.y = W[(size_t)(k + khalf + 1) * HID_ + n0 + r];
        acc = __builtin_amdgcn_wmma_f32_16x16x4_f32(false, a, false, b,
                                                    (short)0, acc, false, false);
    }
    const int col   = n0 + r;
    const int mbase = m0 + ((lane >> 4) * 8);
    #pragma unroll
    for (int rr = 0; rr < 8; ++rr)
        C[(size_t)(mbase + rr) * HID_ + col] = acc[rr];
}

// ---------------------------------------------------------------------------
// Kernel 2: activations. One block (256 thr) per token; wave = head (D==32).
// ---------------------------------------------------------------------------
__global__ void activate_kernel(const float* __restrict__ X,
                                const float* __restrict__ Act,
                                const float* __restrict__ Wbeta,
                                const float* __restrict__ Walpha,
                                const float* __restrict__ Wku,
                                const float* __restrict__ Wvu,
                                const float* __restrict__ Wgamma,
                                float* __restrict__ qs,
                                float* __restrict__ ks,
                                float* __restrict__ ku,
                                float* __restrict__ vu,
                                float* __restrict__ beta,
                                float* __restrict__ alpha,
                                float* __restrict__ gamma)
{
    const int t = blockIdx.x;
    const int h = threadIdx.x >> 5;
    const int d = threadIdx.x & 31;
    const size_t vbase = (size_t)t * HID_ + h * D_ + d;

    float q = qs[vbase];
    q = q * sigmoidf_(q);
    float ssq = waveAllSum(q * q);
    qs[vbase] = q / sqrtf(ssq + EPS_);

    float k = ks[vbase];
    k = k * sigmoidf_(k);
    float ssk = waveAllSum(k * k);
    ks[vbase] = k / sqrtf(ssk + EPS_);

    const float* act = Act + (size_t)t * NA_;
    float kur = 0.0f, vur = 0.0f;
    #pragma unroll
    for (int a = 0; a < NA_; ++a) {
        const float av = act[a];
        kur += av * Wku[(size_t)a * HID_ + h * D_ + d];
        vur += av * Wvu[(size_t)a * HID_ + h * D_ + d];
    }
    kur = kur * sigmoidf_(kur);
    float ssu = waveAllSum(kur * kur);
    ku[vbase] = kur / sqrtf(ssu + EPS_);
    vu[vbase] = vur;

    const float* xr = X + (size_t)t * F_;
    float pb = 0.0f, pa = 0.0f;
    for (int f = d; f < F_; f += 32) {
        const float xv = xr[f];
        pb += xv * Wbeta [(size_t)f * H_ + h];
        pa += xv * Walpha[(size_t)f * H_ + h];
    }
    pb = waveAllSum(pb);
    pa = waveAllSum(pa);

    float pg = (d < NA_) ? act[d] * Wgamma[(size_t)d * H_ + h] : 0.0f;
    pg = waveAllSum(pg);

    if (d == 0) {
        const size_t sh = (size_t)t * H_ + h;
        beta [sh] = sigmoidf_(pb);
        alpha[sh] = sigmoidf_(pa);
        gamma[sh] = sigmoidf_(pg);
    }
}

// ---------------------------------------------------------------------------
// Kernel 3: serial scan with async double-buffered LDS prefetch.
// grid = B*H = 32 single-wave blocks; lane i = row of 32x32 state matrix.
//   H <- A(1-m)(H - beta (Hk) k^T) + Bg vu ku^T + (beta vs - Bg beta c vu) k^T
//   out = H q ,  c = ku . ks
// ---------------------------------------------------------------------------
__device__ __forceinline__ void issue_chunk(const float* ks, const float* ku,
                                            const float* qs, const float* vs,
                                            const float* vu,
                                            const float* beta, const float* alpha,
                                            const float* gamma, const float* mask,
                                            int t0, int h, unsigned ldsBase, int lane)
{
    const int stepIn = lane >> 3;            // step within group of 4
    const int quart  = lane & 7;             // 16B sub-chunk of 32-float row
    const float* bases[5] = { ks, ku, qs, vs, vu };
    #pragma unroll
    for (int st = 0; st < 5; ++st) {
        #pragma unroll
        for (int o = 0; o < 4; ++o) {
            const int step = o * 4 + stepIn;
            unsigned long long g = (unsigned long long)(uintptr_t)
                (bases[st] + (size_t)(t0 + step) * HID_ + h * D_ + quart * 4);
            unsigned l = ldsBase +
                (unsigned)((st * T_ * D_ + step * D_ + quart * 4) * 4);
            ASYNC_B128(l, g);
        }
    }
    // scalar streams: lanes 0-15 / 16-31 carry two streams per instruction
    const int tl = lane & 15;
    {
        const float* p = (lane < 16)
            ? (beta  + (size_t)(t0 + tl) * H_ + h)
            : (alpha + (size_t)(t0 + tl) * H_ + h);
        unsigned l = ldsBase +
            (unsigned)((OFF_SC + (lane < 16 ? 0 : T_) + tl) * 4);
        ASYNC_B32(l, (unsigned long long)(uintptr_t)p);
    }
    {
        const float* p = (lane < 16)
            ? (gamma + (size_t)(t0 + tl) * H_ + h)
            : (mask  + (size_t)(t0 + tl));
        unsigned l = ldsBase +
            (unsigned)((OFF_SC + (lane < 16 ? 2 * T_ : 3 * T_) + tl) * 4);
        ASYNC_B32(l, (unsigned long long)(uintptr_t)p);
    }
}

__global__ void scan_kernel(const float* __restrict__ qs,
                            const float* __restrict__ ks,
                            const float* __restrict__ vs,
                            const float* __restrict__ ku,
                            const float* __restrict__ vu,
                            const float* __restrict__ beta,
                            const float* __restrict__ alpha,
                            const float* __restrict__ gamma,
                            const float* __restrict__ mask,
                            const float* __restrict__ carry,
                            float* __restrict__ outv,
                            float* __restrict__ newCarry)
{
    __shared__ __align__(16) float sh[2][BUF_F];

    const int bh   = blockIdx.x;     // b*H + h
    const int b    = bh >> 3;
    const int h    = bh & 7;
    const int lane = threadIdx.x;    // row i of the state matrix

    // LDS byte offsets of the two buffers (low 32 bits of generic address)
    const unsigned base0 = (unsigned)(uintptr_t)&sh[0][0];
    const unsigned base1 = (unsigned)(uintptr_t)&sh[1][0];

    float Hrow[D_];
    #pragma unroll
    for (int j = 0; j < D_; ++j)
        Hrow[j] = carry[(size_t)bh * D_ * D_ + (size_t)lane * D_ + j];

    const int tok0 = b * S_;
    issue_chunk(ks, ku, qs, vs, vu, beta, alpha, gamma, mask, tok0,           h, base0, lane);
    issue_chunk(ks, ku, qs, vs, vu, beta, alpha, gamma, mask, tok0 + T_,      h, base1, lane);

    for (int c = 0; c < NCHUNK; ++c) {
        if (c + 1 < NCHUNK) WAIT_ASYNC(OPS_PER_CHUNK);  // chunk c landed
        else                WAIT_ASYNC(0);              // last chunk: drain

        const float* buf = sh[c & 1];
        const int    t0  = tok0 + c * T_;

        for (int s = 0; s < T_; ++s) {
            // broadcast vectors: uniform-address LDS reads (no shuffles)
            float kA[D_], kuA[D_], qA[D_];
            #pragma unroll
            for (int j4 = 0; j4 < 8; ++j4) {
                const float4 a = *(const float4*)(buf + OFF_K  + s * D_ + j4 * 4);
                const float4 u = *(const float4*)(buf + OFF_KU + s * D_ + j4 * 4);
                const float4 w = *(const float4*)(buf + OFF_Q  + s * D_ + j4 * 4);
                kA [j4*4+0] = a.x; kA [j4*4+1] = a.y; kA [j4*4+2] = a.z; kA [j4*4+3] = a.w;
                kuA[j4*4+0] = u.x; kuA[j4*4+1] = u.y; kuA[j4*4+2] = u.z; kuA[j4*4+3] = u.w;
                qA [j4*4+0] = w.x; qA [j4*4+1] = w.y; qA [j4*4+2] = w.z; qA [j4*4+3] = w.w;
            }
            const float vL  = buf[OFF_V  + s * D_ + lane];
            const float vuL = buf[OFF_VU + s * D_ + lane];
            const float bet = buf[OFF_SC + 0 * T_ + s];
            const float dec = buf[OFF_SC + 1 * T_ + s] *
                              (1.0f - buf[OFF_SC + 3 * T_ + s]);
            const float Bg  = buf[OFF_SC + 2 * T_ + s];

            // c = ku.ks (every lane has both full vectors) and Hk_i
            float cdot = 0.0f, Hk = 0.0f;
            #pragma unroll
            for (int j = 0; j < D_; ++j) {
                cdot += kuA[j] * kA[j];
                Hk   += Hrow[j] * kA[j];
            }

            const float di = -bet * Hk;
            const float gi = Bg * vuL;
            const float wi = bet * vL - Bg * bet * cdot * vuL;

            float o = 0.0f;
            #pragma unroll
            for (int j = 0; j < D_; ++j) {
                const float hn = dec * (Hrow[j] + di * kA[j])
                               + gi * kuA[j] + wi * kA[j];
                Hrow[j] = hn;
                o += hn * qA[j];
            }
            outv[(size_t)(t0 + s) * HID_ + h * D_ + lane] = o;
        }

        if (c + 2 < NCHUNK) {
            WAIT_DS0();   // all DS reads of this buffer retired before refill
            issue_chunk(ks, ku, qs, vs, vu, beta, alpha, gamma, mask,
                        tok0 + (c + 2) * T_, h, (c & 1) ? base1 : base0, lane);
        }
    }

    #pragma unroll
    for (int j = 0; j < D_; ++j)
        newCarry[(size_t)bh * D_ * D_ + (size_t)lane * D_ + j] = Hrow[j];
}

// ---------------------------------------------------------------------------
// Kernel 4: RMSNorm over HIDDEN=256, in place, * rms_scale.
// ---------------------------------------------------------------------------
__global__ void rms_kernel(float* __restrict__ outv,
                           const float* __restrict__ scale)
{
    __shared__ float red[8];
    const int t = blockIdx.x;
    const int c = threadIdx.x;
    const float v = outv[(size_t)t * HID_ + c];
    const float p = waveAllSum(v * v);
    const int wave = c >> 5, lane = c & 31;
    if (lane == 0) red[wave] = p;
    __syncthreads();
    float tot = 0.0f;
    #pragma unroll
    for (int w = 0; w < 8; ++w) tot += red[w];
    const float r = rsqrtf(tot * (1.0f / HID_) + EPS_);
    outv[(size_t)t * HID_ + c] = v * r * scale[c];
}

// ---------------------------------------------------------------------------
// Kernel 5: output GEMM  Y = On (8192x256) @ Wout (256x512) + bout
// ---------------------------------------------------------------------------
__global__ void out_gemm(const float* __restrict__ On,
                         const float* __restrict__ Wout,
                         const float* __restrict__ bout,
                         float* __restrict__ Y)
{
    const int wave = threadIdx.x >> 5;
    const int lane = threadIdx.x & 31;
    const int m0   = blockIdx.x * 16;
    const int n0   = (blockIdx.y * 8 + wave) * 16;

    const int r     = lane & 15;
    const int khalf = (lane >> 4) * 2;

    v8f acc = {};
    const float* Orow = On + (size_t)(m0 + r) * HID_;
    for (int k = 0; k < HID_; k += 4) {
        v2f a, b;
        a.x = Orow[k + khalf];
        a.y = Orow[k + khalf + 1];
        b.x = Wout[(size_t)(k + khalf)     * F_ + n0 + r];
        b.y = Wout[(size_t)(k + khalf + 1) * F_ + n0 + r];
        acc = __builtin_amdgcn_wmma_f32_16x16x4_f32(false, a, false, b,
                                                    (short)0, acc, false, false);
    }
    const int   col   = n0 + r;
    const float bias  = bout[col];
    const int   mbase = m0 + ((lane >> 4) * 8);
    #pragma unroll
    for (int rr = 0; rr < 8; ++rr)
        Y[(size_t)(mbase + rr) * F_ + col] = acc[rr] + bias;
}

// ---------------------------------------------------------------------------
// Launch
// ---------------------------------------------------------------------------
extern "C" void kernel_launch(void* const* d_in, const int* in_sizes, int n_in,
                              void* d_out, int out_size, void* d_ws, size_t ws_size,
                              hipStream_t stream)
{
    const float* x      = (const float*)d_in[0];
    const float* action = (const float*)d_in[1];
    const float* mask   = (const float*)d_in[2];
    const float* carry  = (const float*)d_in[3];
    const float* Wq     = (const float*)d_in[4];
    const float* Wk     = (const float*)d_in[5];
    const float* Wv     = (const float*)d_in[6];
    const float* Wbeta  = (const float*)d_in[7];
    const float* Walpha = (const float*)d_in[8];
    const float* Wku    = (const float*)d_in[9];
    const float* Wvu    = (const float*)d_in[10];
    const float* Wgamma = (const float*)d_in[11];
    const float* rscale = (const float*)d_in[12];
    const float* Wout   = (const float*)d_in[13];
    const float* bout   = (const float*)d_in[14];

    float* newCarry = (float*)d_out;
    float* y        = (float*)d_out + (size_t)B_ * H_ * D_ * D_;

    const size_t TOKV = (size_t)NTOK * HID_;
    const size_t TOKH = (size_t)NTOK * H_;
    float* ws    = (float*)d_ws;
    float* qs    = ws;
    float* ks    = qs    + TOKV;
    float* vs    = ks    + TOKV;
    float* kuW   = vs    + TOKV;
    float* vuW   = kuW   + TOKV;
    float* outv  = vuW   + TOKV;
    float* betaW = outv  + TOKV;
    float* alphW = betaW + TOKH;
    float* gamW  = alphW + TOKH;
    (void)ws_size; (void)in_sizes; (void)n_in; (void)out_size;

    proj_gemm<<<dim3(NTOK / 16, 6), 256, 0, stream>>>(x, Wq, Wk, Wv, qs, ks, vs);

    activate_kernel<<<NTOK, 256, 0, stream>>>(x, action, Wbeta, Walpha,
                                              Wku, Wvu, Wgamma,
                                              qs, ks, kuW, vuW,
                                              betaW, alphW, gamW);

    scan_kernel<<<B_ * H_, D_, 0, stream>>>(qs, ks, vs, kuW, vuW,
                                            betaW, alphW, gamW, mask, carry,
                                            outv, newCarry);

    rms_kernel<<<NTOK, HID_, 0, stream>>>(outv, rscale);

    out_gemm<<<dim3(NTOK / 16, 4), 256, 0, stream>>>(outv, Wout, bout, y);
}